// PatchNCELoss_43525198577925
// MI455X (gfx1250) — compile-verified
//
#include <hip/hip_runtime.h>
#include <hip/hip_bf16.h>
#include <math.h>

typedef __attribute__((ext_vector_type(2))) float v2f;
typedef __attribute__((ext_vector_type(8))) float v8f;

namespace {
constexpr int   kB      = 16;
constexpr int   kC      = 128;
constexpr int   kN      = 2048;
constexpr int   kNT     = kN / 16;     // 128 16-wide tiles along m (and n)
constexpr float kInvT   = 10.0f;       // 1 / TEMPERATURE
}

// One wave per (b, 16-row n-tile). Computes logits row-block against all m,
// online-softmax per row, folds in the positive logit, writes one partial
// loss sum (16 rows) per block into `partial`.
__global__ __launch_bounds__(32)
void patchnce_main_kernel(const float* __restrict__ q,
                          const float* __restrict__ pos,
                          const float* __restrict__ neg,
                          float* __restrict__ partial) {
  const int lane = threadIdx.x & 31;
  const int h    = lane >> 4;   // half: 0 or 1
  const int t    = lane & 15;   // position within half
  const int b    = blockIdx.y;
  const int n0   = blockIdx.x << 4;

  const float* qb = q   + (size_t)b * kC * kN;
  const float* pb = pos + (size_t)b * kC * kN;
  const float* gb = neg + (size_t)b * kC * kN;

  // ---- l_positive: lane computes row t over c in [h*64, h*64+64), halves merged.
  float lp = 0.0f;
  {
    const float* qp = qb + (size_t)(h * 64) * kN + n0 + t;
    const float* pp = pb + (size_t)(h * 64) * kN + n0 + t;
#pragma unroll 8
    for (int c = 0; c < 64; ++c)
      lp = fmaf(qp[(size_t)c * kN], pp[(size_t)c * kN], lp);
  }
  lp += __shfl_xor(lp, 16);   // every lane with (lane&15)==r now holds lpos[r]

  // ---- A panel (16 x 128 of Q) resident in VGPRs.
  // WMMA f32 16x16x4 A layout: lane -> M = lane&15; VGPR j -> K = 2*(lane>>4)+j.
  v2f apan[32];
  {
    const float* ap = qb + (size_t)(2 * h) * kN + n0 + t;
#pragma unroll
    for (int kk = 0; kk < 32; ++kk) {
      apan[kk].x = ap[(size_t)(4 * kk + 0) * kN];
      apan[kk].y = ap[(size_t)(4 * kk + 1) * kN];
    }
  }

  // Online softmax state: VGPR i covers row (i + 8*h), this lane's columns.
  float mx[8], sm[8];
#pragma unroll
  for (int i = 0; i < 8; ++i) { mx[i] = -INFINITY; sm[i] = 0.0f; }

  const float* bbase = gb + (size_t)(2 * h) * kN + t;

  for (int mt = 0; mt < kNT; ++mt) {
    // ---- B tile (128 x 16 of Neg), same lane/K striping as A.
    const float* bp = bbase + (mt << 4);
    v2f btile[32];
#pragma unroll
    for (int kk = 0; kk < 32; ++kk) {
      btile[kk].x = bp[(size_t)(4 * kk + 0) * kN];
      btile[kk].y = bp[(size_t)(4 * kk + 1) * kN];
    }

    // ---- 32 x V_WMMA_F32_16X16X4_F32 over K=128, 4 accumulators to
    // shorten the D->C dependency chain.
    v8f acc0 = {}, acc1 = {}, acc2 = {}, acc3 = {};
#pragma unroll
    for (int kk = 0; kk < 32; kk += 4) {
      acc0 = __builtin_amdgcn_wmma_f32_16x16x4_f32(false, apan[kk + 0], false, btile[kk + 0],
                                                   (short)0, acc0, false, false);
      acc1 = __builtin_amdgcn_wmma_f32_16x16x4_f32(false, apan[kk + 1], false, btile[kk + 1],
                                                   (short)0, acc1, false, false);
      acc2 = __builtin_amdgcn_wmma_f32_16x16x4_f32(false, apan[kk + 2], false, btile[kk + 2],
                                                   (short)0, acc2, false, false);
      acc3 = __builtin_amdgcn_wmma_f32_16x16x4_f32(false, apan[kk + 3], false, btile[kk + 3],
                                                   (short)0, acc3, false, false);
    }

    // ---- online (max, sumexp) update; logits scaled by 1/T.
#pragma unroll
    for (int i = 0; i < 8; ++i) {
      float v  = ((acc0[i] + acc1[i]) + (acc2[i] + acc3[i])) * kInvT;
      float m2 = fmaxf(mx[i], v);
      sm[i] = sm[i] * __expf(mx[i] - m2) + __expf(v - m2);
      mx[i] = m2;
    }
  }

  // ---- merge the 16 lanes of each half (each holds 1/16 of a row's columns).
#pragma unroll
  for (int i = 0; i < 8; ++i) {
#pragma unroll
    for (int d = 8; d >= 1; d >>= 1) {
      float om = __shfl_xor(mx[i], d);
      float os = __shfl_xor(sm[i], d);
      float m2 = fmaxf(mx[i], om);
      sm[i] = sm[i] * __expf(mx[i] - m2) + os * __expf(om - m2);
      mx[i] = m2;
    }
  }

  // ---- fold in positive logit (class 0), loss = logsumexp - logit0.
  float wsum = 0.0f;
#pragma unroll
  for (int i = 0; i < 8; ++i) {
    float l0 = __shfl(lp, i + 8 * h) * kInvT;
    float m2 = fmaxf(mx[i], l0);
    float s2 = sm[i] * __expf(mx[i] - m2) + __expf(l0 - m2);
    wsum += __logf(s2) + m2 - l0;   // identical across the 16 lanes of the half
  }
  wsum += __shfl_xor(wsum, 16);     // combine rows 0..7 with rows 8..15
  if (lane == 0)
    partial[(size_t)b * kNT + blockIdx.x] = wsum;
}

// Deterministic fixed-order final reduction: 2048 partials -> mean scalar.
__global__ __launch_bounds__(256)
void patchnce_reduce_kernel(const float* __restrict__ partial,
                            float* __restrict__ out, int n, float scale) {
  __shared__ float sdata[256];
  const int tid = threadIdx.x;
  float s = 0.0f;
  for (int i = tid; i < n; i += 256) s += partial[i];
  sdata[tid] = s;
  __syncthreads();
#pragma unroll
  for (int stride = 128; stride > 0; stride >>= 1) {
    if (tid < stride) sdata[tid] += sdata[tid + stride];
    __syncthreads();
  }
  if (tid == 0) out[0] = sdata[0] * scale;
}

extern "C" void kernel_launch(void* const* d_in, const int* in_sizes, int n_in,
                              void* d_out, int out_size, void* d_ws, size_t ws_size,
                              hipStream_t stream) {
  (void)in_sizes; (void)n_in; (void)out_size; (void)ws_size;
  const float* q   = (const float*)d_in[0];
  const float* pos = (const float*)d_in[1];
  const float* neg = (const float*)d_in[2];
  float* out       = (float*)d_out;
  float* partial   = (float*)d_ws;   // kB * kNT = 2048 floats

  dim3 grid(kNT, kB);
  patchnce_main_kernel<<<grid, 32, 0, stream>>>(q, pos, neg, partial);
  patchnce_reduce_kernel<<<1, 256, 0, stream>>>(partial, out, kB * kNT,
                                                1.0f / (float)(kB * kN));
}